// VectorQuantizer2_32074815767040
// MI455X (gfx1250) — compile-verified
//
#include <hip/hip_runtime.h>

// ---------------------------------------------------------------------------
// VQ-VAE vector quantizer for MI455X (gfx1250, wave32, WMMA).
//
// z   : [8, 8, 32, 32, 32] f32   (b, c, l, h, w), c == e_dim == 8
// emb : [1024, 8] f32
// out : [ z_q (2097152) | loss (1) | unique (1) | idx (262144) ]  (f32)
//
// Score s(n, e) = z_n . emb_e - 0.5*||emb_e||^2 ; argmax(s) == argmin(dist).
// Computed as 16x16 tiles with V_WMMA_F32_16X16X4_F32 (two K=4 steps for K=8),
// with -0.5*||e||^2 preloaded into the WMMA accumulator.  The code-tile sweep
// is unrolled x4 so four independent WMMA accumulator chains are in flight,
// hiding fp32-WMMA latency behind the VALU argmax blocks of neighboring tiles.
// ---------------------------------------------------------------------------

typedef __attribute__((ext_vector_type(2))) float v2f;
typedef __attribute__((ext_vector_type(8))) float v8f;

#define NE        1024                 // codebook size
#define EDIM      8
#define S_SP      32768                // 32*32*32 spatial per batch
#define NB        8
#define NROWS     (NB * S_SP)          // 262144
#define ZQ_ELEMS  (NROWS * EDIM)       // 2097152
#define LOSS_OFF  ZQ_ELEMS
#define UNIQ_OFF  (ZQ_ELEMS + 1)
#define IDX_OFF   (ZQ_ELEMS + 2)

__global__ __launch_bounds__(256) void vq_init_ws(double* loss_acc, int* flags) {
    int t = blockIdx.x * blockDim.x + threadIdx.x;
    if (t == 0) *loss_acc = 0.0;
    if (t < NE) flags[t] = 0;
}

__global__ __launch_bounds__(256) void vq_main(const float* __restrict__ z,
                                               const float* __restrict__ emb,
                                               float* __restrict__ out,
                                               double* __restrict__ loss_acc,
                                               int* __restrict__ flags) {
    __shared__ float s_emb[NE * EDIM];   // row-major codebook, 32 KB
    __shared__ float s_m[NE];            // -0.5 * ||emb||^2
    __shared__ int   s_idx[8 * 16];      // per-wave winning code per row

    const int tid = threadIdx.x;

    // Stage codebook into LDS (coalesced; 8192 floats / 256 threads).
    for (int i = tid; i < NE * EDIM; i += 256) s_emb[i] = emb[i];
    __syncthreads();
    for (int n = tid; n < NE; n += 256) {
        float s = 0.f;
#pragma unroll
        for (int c = 0; c < EDIM; ++c) { float e = s_emb[n * EDIM + c]; s += e * e; }
        s_m[n] = -0.5f * s;
    }
    __syncthreads();

    const int wave  = tid >> 5;
    const int lane  = tid & 31;
    const int laneq = lane & 15;   // position within 16-lane half
    const int hf    = lane >> 4;   // which half of the wave

    const int tile = blockIdx.x * 8 + wave;   // 0 .. 16383 (exact, no tail)
    const int m0   = tile * 16;               // first row of this wave's tile
    const int bq   = m0 >> 15;                // batch index (tiles never straddle)
    const int sq   = m0 & (S_SP - 1);         // spatial base within batch

    // --- A tile: 16 rows x 8 dims of z, split into two 16x4 fp32 A operands.
    // A layout (16x4 f32): lanes 0-15 hold M=0..15 with K=0 (VGPR0), K=1 (VGPR1);
    // lanes 16-31 hold K=2, K=3.  z[b][c][s] flat = (b*8+c)*32768 + s.
    const int   c0 = hf * 2;
    const size_t zrow = (size_t)bq * EDIM * S_SP + (size_t)sq + (size_t)laneq;
    v2f a01, a45;
    a01.x = z[zrow + (size_t)(c0 + 0) * S_SP];
    a01.y = z[zrow + (size_t)(c0 + 1) * S_SP];
    a45.x = z[zrow + (size_t)(c0 + 4) * S_SP];
    a45.y = z[zrow + (size_t)(c0 + 5) * S_SP];

    float best[8];
    int   bidx[8];
#pragma unroll
    for (int j = 0; j < 8; ++j) { best[j] = -3.402823466e38f; bidx[j] = 0; }

    // --- Sweep the 64 codebook tiles; x4 unroll = 4 independent WMMA chains.
#pragma unroll 4
    for (int t = 0; t < 64; ++t) {
        const int code = t * 16 + laneq;          // this lane's column code id
        // B layout (4x16 f32): VGPR0 = K row {0 | 2}, VGPR1 = K row {1 | 3}
        // for lane halves {0-15 | 16-31}.  emb rows are contiguous -> b64 loads.
        const float2 e01 = *(const float2*)&s_emb[code * EDIM + hf * 2];
        const float2 e45 = *(const float2*)&s_emb[code * EDIM + 4 + hf * 2];
        v2f b01; b01.x = e01.x; b01.y = e01.y;
        v2f b45; b45.x = e45.x; b45.y = e45.y;

        const float mm = s_m[code];               // -0.5*||e||^2, broadcast down col
        v8f acc = { mm, mm, mm, mm, mm, mm, mm, mm };
        acc = __builtin_amdgcn_wmma_f32_16x16x4_f32(false, a01, false, b01,
                                                    (short)0, acc, false, false);
        acc = __builtin_amdgcn_wmma_f32_16x16x4_f32(false, a45, false, b45,
                                                    (short)0, acc, false, false);
#pragma unroll
        for (int j = 0; j < 8; ++j) {
            float sj = acc[j];
            if (sj > best[j]) { best[j] = sj; bidx[j] = code; }  // strict > keeps lowest idx
        }
    }

    // --- Cross-lane argmax within each 16-lane half (rows live per-VGPR).
#pragma unroll
    for (int j = 0; j < 8; ++j) {
#pragma unroll
        for (int m = 1; m < 16; m <<= 1) {
            float ov = __shfl_xor(best[j], m, 32);
            int   oi = __shfl_xor(bidx[j], m, 32);
            if (ov > best[j] || (ov == best[j] && oi < bidx[j])) { best[j] = ov; bidx[j] = oi; }
        }
    }

    // Stage winners: lower half holds rows m0+j, upper half rows m0+8+j.
#pragma unroll
    for (int j = 0; j < 8; ++j)
        if (laneq == j) s_idx[wave * 16 + hf * 8 + j] = bidx[j];
    __syncthreads();

    const int ridx = s_idx[wave * 16 + laneq];    // winner for row m0+laneq

    // --- Gather z_q, write transposed output (16 contiguous floats per c-plane),
    //     and accumulate the loss partial: loss = 1.25 * mean((z_q - z)^2).
    float part = 0.f;
#pragma unroll
    for (int it = 0; it < 4; ++it) {
        const int c = it * 2 + hf;
        const size_t off = (size_t)(bq * EDIM + c) * S_SP + (size_t)sq + (size_t)laneq;
        const float e  = s_emb[ridx * EDIM + c];
        const float zv = z[off];
        const float d  = e - zv;
        part += d * d;
        out[off] = e;
    }
    if (hf == 0) {
        out[IDX_OFF + m0 + laneq] = (float)ridx;
        flags[ridx] = 1;                          // racy "1" stores are fine
    }
#pragma unroll
    for (int m = 1; m < 32; m <<= 1) part += __shfl_xor(part, m, 32);
    if (lane == 0) atomicAdd(loss_acc, (double)part);
}

__global__ __launch_bounds__(1024) void vq_finalize(const int* __restrict__ flags,
                                                    const double* __restrict__ loss_acc,
                                                    float* __restrict__ out) {
    __shared__ int red[32];
    const int t = threadIdx.x;
    int f = flags[t] ? 1 : 0;
#pragma unroll
    for (int m = 1; m < 32; m <<= 1) f += __shfl_xor(f, m, 32);
    if ((t & 31) == 0) red[t >> 5] = f;
    __syncthreads();
    if (t < 32) {
        int v = red[t];
#pragma unroll
        for (int m = 1; m < 32; m <<= 1) v += __shfl_xor(v, m, 32);
        if (t == 0) {
            out[UNIQ_OFF] = (float)v;
            out[LOSS_OFF] = (float)(1.25 * (*loss_acc) / (double)ZQ_ELEMS);
        }
    }
}

extern "C" void kernel_launch(void* const* d_in, const int* in_sizes, int n_in,
                              void* d_out, int out_size, void* d_ws, size_t ws_size,
                              hipStream_t stream) {
    (void)in_sizes; (void)n_in; (void)out_size; (void)ws_size;
    const float* z   = (const float*)d_in[0];
    const float* emb = (const float*)d_in[1];
    float* out = (float*)d_out;

    double* loss_acc = (double*)d_ws;                 // 8 bytes
    int*    flags    = (int*)((char*)d_ws + 8);       // 1024 ints

    vq_init_ws<<<4, 256, 0, stream>>>(loss_acc, flags);
    vq_main<<<NROWS / (16 * 8), 256, 0, stream>>>(z, emb, out, loss_acc, flags);
    vq_finalize<<<1, 1024, 0, stream>>>(flags, loss_acc, out);
}